// MaskedWeight_50397146251529
// MI455X (gfx1250) — compile-verified
//
#include <hip/hip_runtime.h>

// MaskedWeight fused kernel for gfx1250 (MI455X).
// One workgroup (8 wave32) per sample. Reads only the 136 lower+diagonal
// 32x32 blocks of hyper_w (143MB instead of 268MB -> ~6us HBM floor).
// Matvec done with v_wmma_f32_16x16x4_f32, accumulating in the C operand
// across the block-row (B operand = x chunk broadcast over all 16 columns,
// so every column of D holds the same partial y).

typedef float v2f __attribute__((ext_vector_type(2)));
typedef float v8f __attribute__((ext_vector_type(8)));

#define TS 36   // LDS tile row stride in floats (16B aligned, bank-conflict friendly)

__global__ __launch_bounds__(256) void mw_fused_kernel(
    const float* __restrict__ x,      // [256,512]
    const float* __restrict__ hw,     // [256,512,512]
    const float* __restrict__ hb,     // [256,512]
    const float* __restrict__ cumul,  // [256,16,32,1] -> flat [256,512]
    const float* __restrict__ sfp,    // [1]
    float* __restrict__ out)          // [0..131071]=outputs, [131072..]=out_log_det
{
    __shared__ float xs[512];
    __shared__ float ylds[512];
    __shared__ float cum_s[512];
    __shared__ float tiles[8][32 * TS];
    __shared__ float f2s;

    const int b    = blockIdx.x;
    const int t    = threadIdx.x;
    const int lane = t & 31;
    const int wv   = t >> 5;
    const int half = lane >> 4;      // 0: lanes 0-15, 1: lanes 16-31
    const int r15  = lane & 15;
    const float sf = sfp[0];

    // Stage x, init reduction scalar.
    xs[t]       = x[(size_t)b * 512 + t];
    xs[t + 256] = x[(size_t)b * 512 + 256 + t];
    if (t == 0) f2s = 0.0f;
    __syncthreads();

    float* tile = tiles[wv];
    const float* hwb = hw + (size_t)b * 512 * 512;

    float f2acc = 0.0f;

    // Wave wv handles block-rows {wv, 15-wv}: (wv+1)+(16-wv) = 17 blocks each.
    for (int ri = 0; ri < 2; ++ri) {
        const int rb = ri ? (15 - wv) : wv;
        v8f Ca = {0.f, 0.f, 0.f, 0.f, 0.f, 0.f, 0.f, 0.f}; // rows rb*32+0..15
        v8f Cb = {0.f, 0.f, 0.f, 0.f, 0.f, 0.f, 0.f, 0.f}; // rows rb*32+16..31

        for (int cb = 0; cb <= rb; ++cb) {
            const float* bp = hwb + (size_t)(rb * 32) * 512 + cb * 32;
            const bool isdiag = (cb == rb);

            // Load 32x32 block coalesced (float4/lane x8), apply mask op,
            // accumulate w^2, stage into padded LDS tile.
            #pragma unroll
            for (int j = 0; j < 8; ++j) {
                const int e   = j * 32 + lane;       // 0..255 float4 slots
                const int row = e >> 3;              // 0..31
                const int col = (lane & 7) << 2;     // 0,4,...,28
                float4 v = *(const float4*)(bp + (size_t)row * 512 + col);
                if (isdiag) {
                    v.x = expf(v.x); v.y = expf(v.y);
                    v.z = expf(v.z); v.w = expf(v.w);
                }
                f2acc += v.x * v.x + v.y * v.y + v.z * v.z + v.w * v.w;
                *(float4*)&tile[row * TS + col] = v;
            }
            // Wave-local LDS RAW: DS ops are in-order per wave; compiler
            // inserts s_wait_dscnt as needed.

            // Matvec via WMMA f32 16x16x4: A = w chunk, B = x chunk broadcast
            // across all N columns, C accumulates y across cb and kc.
            #pragma unroll
            for (int kc = 0; kc < 8; ++kc) {
                const int c2 = kc * 4 + half * 2;
                v2f bb = *(const v2f*)&xs[cb * 32 + c2];               // B: V0={K0|K2}, V1={K1|K3}
                v2f a0 = *(const v2f*)&tile[r15 * TS + c2];            // rows 0..15
                v2f a1 = *(const v2f*)&tile[(16 + r15) * TS + c2];     // rows 16..31
                Ca = __builtin_amdgcn_wmma_f32_16x16x4_f32(
                        false, a0, false, bb, (short)0, Ca, false, false);
                Cb = __builtin_amdgcn_wmma_f32_16x16x4_f32(
                        false, a1, false, bb, (short)0, Cb, false, false);
            }
        }

        // All 16 columns of D are identical; VGPR r holds row r (lanes 0-15)
        // and row r+8 (lanes 16-31). Lane 0 / lane 16 publish y rows.
        if (r15 == 0) {
            const int off = rb * 32 + half * 8;
            #pragma unroll
            for (int r = 0; r < 8; ++r) {
                ylds[off + r]      = Ca[r];
                ylds[off + 16 + r] = Cb[r];
            }
        }
    }

    // Reduce w_F2: intra-wave butterfly, then one LDS float atomic per wave.
    #pragma unroll
    for (int off = 16; off > 0; off >>= 1)
        f2acc += __shfl_xor(f2acc, off, 32);
    if (lane == 0) atomicAdd(&f2s, f2acc);

    // Stage log_grad_cumul while waiting.
    cum_s[t]       = cumul[(size_t)b * 512 + t];
    cum_s[t + 256] = cumul[(size_t)b * 512 + 256 + t];
    __syncthreads();

    const float F2    = f2s;
    const float scale = expf(sf) * rsqrtf(F2);

    // outputs[b,o] = exp(sf)/sqrt(F2) * y[o] + hyper_b[b,o]
    #pragma unroll
    for (int k = 0; k < 2; ++k) {
        const int o = t + k * 256;
        out[(size_t)b * 512 + o] = scale * ylds[o] + hb[(size_t)b * 512 + o];
    }

    // out_log_det[b,d,o] = sf - 0.5*log(F2) + logsumexp_i(diag[d,o,i] + cumul[d,i])
    const float lbias = sf - 0.5f * logf(F2);
    #pragma unroll
    for (int k = 0; k < 2; ++k) {
        const int idx = t + k * 256;      // 0..511
        const int d = idx >> 5;
        const int o = idx & 31;
        const float* rp = hwb + (size_t)(d * 32 + o) * 512 + d * 32;
        float4 rv[8];
        float m = -3.0e38f;
        #pragma unroll
        for (int j = 0; j < 8; ++j) {
            float4 v = *(const float4*)(rp + j * 4);
            v.x += cum_s[d * 32 + j * 4 + 0];
            v.y += cum_s[d * 32 + j * 4 + 1];
            v.z += cum_s[d * 32 + j * 4 + 2];
            v.w += cum_s[d * 32 + j * 4 + 3];
            rv[j] = v;
            m = fmaxf(m, fmaxf(fmaxf(v.x, v.y), fmaxf(v.z, v.w)));
        }
        float s = 0.0f;
        #pragma unroll
        for (int j = 0; j < 8; ++j) {
            s += expf(rv[j].x - m) + expf(rv[j].y - m)
               + expf(rv[j].z - m) + expf(rv[j].w - m);
        }
        out[131072 + (size_t)b * 512 + idx] = lbias + m + logf(s);
    }
}

extern "C" void kernel_launch(void* const* d_in, const int* in_sizes, int n_in,
                              void* d_out, int out_size, void* d_ws, size_t ws_size,
                              hipStream_t stream) {
    const float* x     = (const float*)d_in[0];
    const float* hw    = (const float*)d_in[1];
    const float* hb    = (const float*)d_in[2];
    const float* cumul = (const float*)d_in[3];
    const float* sf    = (const float*)d_in[4];
    float* out = (float*)d_out;

    mw_fused_kernel<<<256, 256, 0, stream>>>(x, hw, hb, cumul, sf, out);
}